// Attention_11467562680552
// MI455X (gfx1250) — compile-verified
//
#include <hip/hip_runtime.h>
#include <cstdint>

// ---------------------------------------------------------------------------
// scores[s] = sigmoid( sum_d E[s,d] * h[d] ),  E: 32768x1024 fp32, h: 1024 fp32
//
// MI455X roofline: 134 MB of reads at 23.3 TB/s => ~5.8 us floor; AI ~= 0.5
// FLOP/byte => pure bandwidth problem. Strategy: stream E once with
// global_load_async_to_lds_b128 (coalesced 512B per instruction, ASYNCcnt
// double-buffered), and feed V_WMMA_F32_16X16X4_F32 from LDS with a
// bank-conflict-free padded layout. WMMA's 16 redundant output columns cost
// nothing (compute is ~1000x below the bandwidth limit) and keep the matrix
// pipe as the reduction engine.
// ---------------------------------------------------------------------------

typedef __attribute__((ext_vector_type(2))) float v2f;
typedef __attribute__((ext_vector_type(8))) float v8f;

#define D_DIM      1024
#define SEQ        32768
#define KT         64                   // k-extent staged per chunk
#define CHUNKS     (D_DIM / KT)         // 16
#define ROW_STRIDE 68                   // KT + 4 pad dwords: rows differ by 4 banks
#define WAVES      4                    // waves per block (wave32)
#define TILE_FLOATS (16 * ROW_STRIDE)   // one 16-row staging buffer

__global__ __launch_bounds__(WAVES * 32, 1)
void attn_score_wmma_kernel(const float* __restrict__ hidden,
                            const float* __restrict__ enc,
                            float* __restrict__ out)
{
    __shared__ float h_lds[D_DIM];
    __shared__ float tiles[WAVES * 2 * TILE_FLOATS]; // double buffer per wave

    const int tid  = threadIdx.x;
    const int wave = tid >> 5;
    const int lane = tid & 31;

    // Stage h (4 KB) once per block, coalesced.
    for (int i = tid; i < D_DIM; i += WAVES * 32) h_lds[i] = hidden[i];
    __syncthreads();

    const int tileIdx  = blockIdx.x * WAVES + wave; // 16-row tile owned by wave
    const int row_base = tileIdx * 16;

    float* tbuf0 = &tiles[(wave * 2 + 0) * TILE_FLOATS];
    float* tbuf1 = &tiles[(wave * 2 + 1) * TILE_FLOATS];

    // Async-copy lane geometry: each instruction moves 512B = two 256B rows.
    const int cp_half = lane >> 4;        // which row of the pair
    const int cp_col  = (lane & 15) * 4;  // float column (16B per lane)

    // Issue one 16x64 fp32 tile (8 x b128 async loads, ASYNCcnt += 8).
    auto issue_chunk = [&](float* buf, int c) {
        const float* gbase = enc + (size_t)row_base * D_DIM + c * KT;
#pragma unroll
        for (int p = 0; p < 8; ++p) {
            const int r = p * 2 + cp_half;
            const float* g = gbase + (size_t)r * D_DIM + cp_col;
            const uint32_t lds_addr =
                (uint32_t)(uintptr_t)(buf + r * ROW_STRIDE + cp_col);
            asm volatile("global_load_async_to_lds_b128 %0, %1, off"
                         :: "v"(lds_addr), "v"((uint64_t)(uintptr_t)g)
                         : "memory");
        }
    };

    v8f acc = {0.f, 0.f, 0.f, 0.f, 0.f, 0.f, 0.f, 0.f};

    // WMMA operand geometry (ISA 16x4 f32 A layout):
    //   lane m    (0-15) : A[m, k+0], A[m, k+1]
    //   lane m+16        : A[m, k+2], A[m, k+3]
    // => one ds_load_b64 per lane; with ROW_STRIDE=68 (68%64==4) the 32 lanes
    //    of a step touch all 64 LDS banks exactly once (conflict-free).
    const int arow = lane & 15;
    const int asel = (lane >> 4) * 2;
    const int bsel = lane >> 4;   // B rows K0+{0,1} in v0, K0+{2,3} in v1

    auto compute_chunk = [&](const float* buf, int c) {
#pragma unroll
        for (int k = 0; k < KT; k += 4) {
            v2f a = *(const v2f*)(buf + arow * ROW_STRIDE + k + asel);
            v2f b;
            b.x = h_lds[c * KT + k + bsel];       // B[k+0/1, n] = h broadcast
            b.y = h_lds[c * KT + k + 2 + bsel];   // B[k+2/3, n]
            acc = __builtin_amdgcn_wmma_f32_16x16x4_f32(
                false, a, false, b, (short)0, acc, false, false);
        }
    };

    // Double-buffered stream over K: prefetch chunk c+1 while computing c.
    issue_chunk(tbuf0, 0);
    for (int c = 0; c < CHUNKS; ++c) {
        if (c + 1 < CHUNKS) {
            issue_chunk((c & 1) ? tbuf0 : tbuf1, c + 1);
            // 16 outstanding; <=8 left means the previous chunk landed.
            asm volatile("s_wait_asynccnt 8" ::: "memory");
        } else {
            asm volatile("s_wait_asynccnt 0" ::: "memory");
        }
        compute_chunk((c & 1) ? tbuf1 : tbuf0, c);
    }

    // D layout: lane holds column n=lane&15; VGPR r = row r + 8*(lane>>4).
    // All columns are identical (B was broadcast) -> column 0 suffices.
    if ((lane & 15) == 0) {
        const int rbase = row_base + (lane >> 4) * 8;
#pragma unroll
        for (int r = 0; r < 8; ++r) {
            const float s = acc[r];
            out[rbase + r] = 1.0f / (1.0f + __expf(-s)); // sigmoid
        }
    }
}

extern "C" void kernel_launch(void* const* d_in, const int* in_sizes, int n_in,
                              void* d_out, int out_size, void* d_ws, size_t ws_size,
                              hipStream_t stream) {
    const float* hidden = (const float*)d_in[0];   // [1024]
    const float* enc    = (const float*)d_in[1];   // [32768, 1024]
    float* out          = (float*)d_out;           // [1,1,32768] flat

    const int blocks = SEQ / (16 * WAVES);         // 512 blocks x 4 waves = 2048 tiles
    attn_score_wmma_kernel<<<blocks, WAVES * 32, 0, stream>>>(hidden, enc, out);
}